// MaxBlurPool2d_67465346286216
// MI455X (gfx1250) — compile-verified
//
#include <hip/hip_runtime.h>

// -------- problem geometry (fixed by the reference) --------
#define H   256
#define W   256
#define OH  128
#define OW  128

// -------- tiling --------
#define TOW 64               // output tile width
#define TOH 16               // output tile height
#define TILES_X (OW / TOW)   // 2
#define TILES_Y (OH / TOH)   // 8
#define TILES_PER_PLANE (TILES_X * TILES_Y)  // 16

#define IN_W  (2*TOW + 5)    // 133  input cols (blur +-2 @ stride2, maxpool +-1)
#define IN_H  (2*TOH + 5)    // 37
#define IN_WP 136            // padded LDS row stride (dwords)
#define M_W   (2*TOW + 3)    // 131  maxpool cols needed by the blur
#define M_H   (2*TOH + 3)    // 35
#define M_WP  132

// -------- CDNA5 async global->LDS path (probe via __has_builtin) --------
#if defined(__has_builtin)
#  if __has_builtin(__builtin_amdgcn_global_load_async_to_lds_b32)
#    define HAVE_ASYNC_LDS 1
#  endif
#  if __has_builtin(__builtin_amdgcn_s_wait_asynccnt)
#    define HAVE_WAIT_ASYNC 1
#  endif
#endif

#define AS1 __attribute__((address_space(1)))
#define AS3 __attribute__((address_space(3)))

__device__ __forceinline__ void async_copy_b32(const float* gsrc, float* ldst) {
#ifdef HAVE_ASYNC_LDS
    __builtin_amdgcn_global_load_async_to_lds_b32((AS1 int*)gsrc, (AS3 int*)ldst, 0, 0);
#else
    *ldst = *gsrc;
#endif
}

__device__ __forceinline__ void async_wait_all() {
#ifdef HAVE_ASYNC_LDS
#  ifdef HAVE_WAIT_ASYNC
    __builtin_amdgcn_s_wait_asynccnt(0);
#  else
    asm volatile("s_wait_asynccnt 0" ::: "memory");
#  endif
#endif
}

__global__ __launch_bounds__(256) void maxblurpool_kernel(
    const float* __restrict__ x, float* __restrict__ out) {

    __shared__ float s_in[IN_H * IN_WP];   // raw input tile, -inf halo
    __shared__ float s_max[M_H * M_WP];    // 3x3 maxpool, 0 for out-of-image

    const int tid   = threadIdx.x;
    const int plane = blockIdx.x / TILES_PER_PLANE;
    const int tile  = blockIdx.x % TILES_PER_PLANE;
    const int tx    = tile % TILES_X;
    const int ty    = tile / TILES_X;
    const int ox0   = tx * TOW;
    const int oy0   = ty * TOH;
    const int ix0   = 2 * ox0 - 3;         // global col of s_in[.][0]
    const int iy0   = 2 * oy0 - 3;         // global row of s_in[0][.]

    const float* __restrict__ xp = x + (size_t)plane * (H * W);
    const float ninf = -__builtin_inff();

    // ---- Phase 0: stage tile.  In-bounds -> async global->LDS copy,
    //      out-of-bounds -> -inf (maxpool padding semantics). ----
    for (int i = tid; i < IN_H * IN_W; i += 256) {
        const int r  = i / IN_W;
        const int c  = i - r * IN_W;
        const int gy = iy0 + r;
        const int gx = ix0 + c;
        if ((unsigned)gx < (unsigned)W && (unsigned)gy < (unsigned)H) {
            async_copy_b32(xp + gy * W + gx, &s_in[r * IN_WP + c]);
        } else {
            s_in[r * IN_WP + c] = ninf;
        }
    }
    async_wait_all();
    __syncthreads();

    // ---- Phase 1: separable 3x3 maxpool, register sliding window over rows.
    //      One thread per maxpool column (131 active).  3 LDS reads + 1 write/elem. ----
    if (tid < M_W) {
        const int c = tid;
        const bool colOK = (unsigned)(ix0 + 1 + c) < (unsigned)W;

        const float* r0 = &s_in[c];
        float h0 = fmaxf(fmaxf(r0[0],        r0[1]),        r0[2]);
        float h1 = fmaxf(fmaxf(r0[IN_WP],    r0[IN_WP + 1]), r0[IN_WP + 2]);
        for (int r = 0; r < M_H; ++r) {
            const float* q = &s_in[(r + 2) * IN_WP + c];
            const float h2 = fmaxf(fmaxf(q[0], q[1]), q[2]);
            const float m  = fmaxf(fmaxf(h0, h1), h2);
            const bool rowOK = (unsigned)(iy0 + 1 + r) < (unsigned)H;
            s_max[r * M_WP + c] = (colOK && rowOK) ? m : 0.0f;
            h0 = h1; h1 = h2;
        }
    }
    __syncthreads();

    // ---- Phase 2: separable 5x5 binomial blur, stride 2.
    //      thread = (out col, row group of 4).  Horizontal taps from LDS
    //      (stride-2 dwords over 32 lanes -> 32 distinct banks, conflict-free),
    //      vertical taps entirely in registers.  Coalesced stores. ----
    const float w0 = 1.0f / 16.0f, w1 = 4.0f / 16.0f, w2 = 6.0f / 16.0f;
    const int ox = tid & 63;               // 0..63
    const int g  = tid >> 6;               // 0..3
    const int y0 = 8 * g;                  // first s_max row this group needs

    float hb[11];
#pragma unroll
    for (int j = 0; j < 11; ++j) {
        const float* q = &s_max[(y0 + j) * M_WP + 2 * ox];
        hb[j] = w0 * (q[0] + q[4]) + w1 * (q[1] + q[3]) + w2 * q[2];
    }

    float* __restrict__ op = out + (size_t)plane * (OH * OW);
#pragma unroll
    for (int k = 0; k < 4; ++k) {
        const int oy = 4 * g + k;          // 0..15
        const float acc = w0 * (hb[2*k] + hb[2*k + 4])
                        + w1 * (hb[2*k + 1] + hb[2*k + 3])
                        + w2 *  hb[2*k + 2];
        op[(size_t)(oy0 + oy) * OW + ox0 + ox] = acc;
    }
}

extern "C" void kernel_launch(void* const* d_in, const int* in_sizes, int n_in,
                              void* d_out, int out_size, void* d_ws, size_t ws_size,
                              hipStream_t stream) {
    const float* x = (const float*)d_in[0];
    float* out = (float*)d_out;
    const int n_planes = in_sizes[0] / (H * W);      // 8 * 128 = 1024
    dim3 grid(n_planes * TILES_PER_PLANE);           // 16384 blocks
    maxblurpool_kernel<<<grid, 256, 0, stream>>>(x, out);
}